// wACSFRad_90958817394879
// MI455X (gfx1250) — compile-verified
//
#include <hip/hip_runtime.h>
#include <hip/hip_bf16.h>
#include <math.h>
#include <stdint.h>

#define NPARAMS   22
#define MAX_TYPES 118
#define EM_FLOATS (MAX_TYPES * NPARAMS * 2)   /* 5192 floats = 20768 B of LDS */
#define CUTOFF_R  8.0f
#define PI_OVER_CUTOFF 0.39269908169872414f   /* pi / 8 */

#if defined(__AMDGCN__) && __has_builtin(__builtin_amdgcn_global_load_async_to_lds_b128)
#define USE_ASYNC_LDS 1
#endif

#if defined(USE_ASYNC_LDS)
typedef int v4i __attribute__((ext_vector_type(4)));
typedef __attribute__((address_space(1))) v4i* glob_v4i_ptr;
typedef __attribute__((address_space(3))) v4i* lds_v4i_ptr;
#endif

// Stage the (eta,mu) table into LDS.
// Async path: each wave issues async global->LDS b128 copies for its own lanes'
// chunks, waits on its *own* ASYNCcnt (per-wave counter), then the workgroup
// barrier guarantees every wave's staged slice is visible in LDS.
__device__ __forceinline__ void stage_table(const float* __restrict__ g,
                                            float* __restrict__ s,
                                            int nfloats)
{
#if defined(USE_ASYNC_LDS)
    int nv4 = nfloats >> 2;
    for (int i = threadIdx.x; i < nv4; i += blockDim.x) {
        glob_v4i_ptr gp = (glob_v4i_ptr)(uintptr_t)(g + 4 * i);
        // low 32 bits of a flat shared address == LDS byte offset
        lds_v4i_ptr  lp = (lds_v4i_ptr)(uint32_t)(uintptr_t)(s + 4 * i);
        __builtin_amdgcn_global_load_async_to_lds_b128(gp, lp, 0, 0);
    }
    // tail (table size not multiple of 4 floats) -- synchronous, thread 0
    if (threadIdx.x == 0) {
        for (int i = nv4 << 2; i < nfloats; ++i) s[i] = g[i];
    }
#if __has_builtin(__builtin_amdgcn_s_wait_asynccnt)
    __builtin_amdgcn_s_wait_asynccnt(0);
#else
    asm volatile("s_wait_asynccnt 0" ::: "memory");
#endif
#else
    for (int i = threadIdx.x; i < nfloats; i += blockDim.x) s[i] = g[i];
#endif
    __syncthreads();
}

extern "C" __global__ __launch_bounds__(256)
void wacsf_edge_kernel(const int* __restrict__ z,
                       const float* __restrict__ xyz,
                       const int* __restrict__ eij,
                       const float* __restrict__ eta_mu,
                       float* __restrict__ out,
                       int n_edges,
                       int em_floats)
{
    __shared__ float s_em[EM_FLOATS];
    stage_table(eta_mu, s_em, em_floats);

    const int2* e2 = (const int2*)eij;
    int stride = (int)(gridDim.x * blockDim.x);
    for (int e = (int)(blockIdx.x * blockDim.x + threadIdx.x); e < n_edges; e += stride) {
        int2 ed  = e2[e];
        int recv = ed.x;
        int send = ed.y;

        float xi0 = xyz[3 * recv + 0];
        float xi1 = xyz[3 * recv + 1];
        float xi2 = xyz[3 * recv + 2];
        float xj0 = xyz[3 * send + 0];
        float xj1 = xyz[3 * send + 1];
        float xj2 = xyz[3 * send + 2];

        float dx = xi0 - xj0;
        float dy = xi1 - xj1;
        float dz = xi2 - xj2;
        float r  = sqrtf(dx * dx + dy * dy + dz * dz);

        // r >= 0, so clip(r,-8,8) == min(r,8)
        float rc = fminf(r, CUTOFF_R);
        float fc = 0.5f * (cosf(rc * PI_OVER_CUTOFF) + 1.0f);
        float w  = (float)z[send];
        float s  = fc * w;

        const float* em = &s_em[z[recv] * (NPARAMS * 2)];
        float* op = out + (size_t)recv * NPARAMS;

#pragma unroll
        for (int p = 0; p < NPARAMS; ++p) {
            float eta = em[2 * p + 0];
            float mu  = em[2 * p + 1];
            float d   = r - mu;
            float g   = expf(-eta * d * d);
            atomicAdd(op + p, s * g);
        }
    }
}

extern "C" void kernel_launch(void* const* d_in, const int* in_sizes, int n_in,
                              void* d_out, int out_size, void* d_ws, size_t ws_size,
                              hipStream_t stream)
{
    (void)n_in; (void)d_ws; (void)ws_size;
    const int*   z      = (const int*)d_in[0];
    const float* xyz    = (const float*)d_in[1];
    const int*   eij    = (const int*)d_in[2];
    const float* eta_mu = (const float*)d_in[3];
    float*       out    = (float*)d_out;

    int n_edges   = in_sizes[2] / 2;
    int em_floats = in_sizes[3];
    if (em_floats > EM_FLOATS) em_floats = EM_FLOATS;

    // segment_sum accumulator must start at zero (d_out is poisoned)
    (void)hipMemsetAsync(d_out, 0, (size_t)out_size * sizeof(float), stream);

    int block = 256;
    int grid  = (n_edges + block - 1) / block;
    if (grid > 4096) grid = 4096;   // cap redundant per-block LDS staging traffic
    if (grid < 1)    grid = 1;

    wacsf_edge_kernel<<<grid, block, 0, stream>>>(z, xyz, eij, eta_mu, out,
                                                  n_edges, em_floats);
}